// WeightedMSELossWithPenalty_74028056314237
// MI455X (gfx1250) — compile-verified
//
#include <hip/hip_runtime.h>
#include <stdint.h>

// WeightedMSELossWithPenalty for MI455X (gfx1250).
// Memory-bound streaming reduction: 336 MB @ 23.3 TB/s ~= 14.4 us floor.
// Uses gfx1250 async global->LDS loads (ASYNCcnt), NT-hinted B128 loads,
// and global_prefetch. WMMA is intentionally not used (no matmul structure).

#define FDIM      21
#define ROWS_PB   1024
#define TPB       256

typedef float v4f __attribute__((ext_vector_type(4)));
typedef int   v4i __attribute__((ext_vector_type(4)));

#if __has_builtin(__builtin_amdgcn_global_load_async_to_lds_b32)
#define ASYNC_BUILTIN 1
#endif

__device__ __forceinline__ void async_wait0() {
#if __has_builtin(__builtin_amdgcn_s_wait_asynccnt)
  __builtin_amdgcn_s_wait_asynccnt(0);
#else
  asm volatile("s_wait_asynccnt 0x0" ::: "memory");
#endif
}

// Copy one dword global -> LDS via the gfx1250 async path (per-lane addresses).
__device__ __forceinline__ void async_b32_to_lds(const int* gsrc, unsigned int* lds_generic) {
  uint32_t loff  = (uint32_t)(uintptr_t)lds_generic;   // LDS aperture: addr[31:0] is the LDS offset
  uint64_t gaddr = (uint64_t)(uintptr_t)gsrc;
#if defined(ASYNC_BUILTIN)
  typedef __attribute__((address_space(1))) int gint_t;   // param is non-const AS1 int*
  typedef __attribute__((address_space(3))) int lint_t;   // LDS side: AS3 int*
  __builtin_amdgcn_global_load_async_to_lds_b32((gint_t*)(uintptr_t)gaddr,
                                                (lint_t*)loff,
                                                /*offset=*/0, /*cpol=*/0);
#else
  asm volatile("global_load_async_to_lds_b32 %0, %1, off"
               :: "v"(loff), "v"(gaddr) : "memory");
#endif
}

__global__ void __launch_bounds__(TPB)
WeightedMSELossWithPenalty_74028056314237_kernel(const float* __restrict__ in,
                                                 const int*   __restrict__ tg,
                                                 const float* __restrict__ wt,
                                                 float*       __restrict__ out,
                                                 int rows)
{
  __shared__ unsigned int s_w[ROWS_PB];   // class value, then bit-cast weight, per row
  const int tid     = threadIdx.x;
  const int rowBase = blockIdx.x * ROWS_PB;

  // ---- Phase A: async DMA the class column (strided gather) into LDS ----
#pragma unroll
  for (int k = 0; k < ROWS_PB / TPB; ++k) {
    int r   = tid + k * TPB;
    int row = rowBase + r;
    if (row < rows)
      async_b32_to_lds(tg + (size_t)row * FDIM + 2, &s_w[r]);
  }
  async_wait0();           // drain ASYNCcnt: this wave's slots are now in LDS

  // class value {100..700} -> weights[class/100 - 1], stored in place as bits
#pragma unroll
  for (int k = 0; k < ROWS_PB / TPB; ++k) {
    int r   = tid + k * TPB;
    int row = rowBase + r;
    float w = 0.0f;
    if (row < rows) {
      unsigned int cls = s_w[r];
      w = wt[cls / 100u - 1u];
    }
    s_w[r] = __float_as_uint(w);
  }
  __syncthreads();

  // ---- Phase B: coalesced NT stream over 1024 rows * 21 cols ----
  const unsigned int total    = (unsigned int)rows * FDIM;
  const unsigned int elemBase = (unsigned int)blockIdx.x * (ROWS_PB * FDIM);
  float accL = 0.0f, accP = 0.0f;

#pragma unroll 1
  for (int it = 0; it < FDIM; ++it) {            // 21 iters * 1024 elems = 21504
    unsigned int i = (unsigned int)it * (TPB * 4) + (unsigned int)tid * 4;
    unsigned int g = elemBase + i;
    if (g + 4 <= total) {
      if (g + 4 * TPB * 4 + 4 <= total) {        // prefetch ~4 iterations ahead
        __builtin_prefetch(in + g + 4 * TPB * 4, 0, 1);
        __builtin_prefetch(tg + g + 4 * TPB * 4, 0, 1);
      }
      v4f x = __builtin_nontemporal_load((const v4f*)(in + g));  // b128 th:NT
      v4i t = __builtin_nontemporal_load((const v4i*)(tg + g));  // b128 th:NT
#pragma unroll
      for (int j = 0; j < 4; ++j) {
        unsigned int idx = i + (unsigned int)j;
        unsigned int r   = idx / 21u;            // mul-shift by compiler
        unsigned int f   = idx - r * 21u;
        float tv = (float)t[j];
        float d  = x[j] - tv;
        accL = __builtin_fmaf(__uint_as_float(s_w[r]) * d, d, accL);
        if (f == 1u) {                           // range penalty on column 1
          float lo = (tv < 0.0f)    ? tv * tv : 0.0f;
          float hv = tv - 1000.0f;
          float hi = (tv > 1000.0f) ? hv * hv : 0.0f;
          accP += lo + hi;
        }
      }
    }
  }

  // ---- Reduce: wave32 shuffle, then cross-wave via LDS, one atomic/block ----
  for (int off = 16; off > 0; off >>= 1) {
    accL += __shfl_down(accL, off, 32);
    accP += __shfl_down(accP, off, 32);
  }
  __shared__ float rL[TPB / 32], rP[TPB / 32];
  int wave = tid >> 5;
  if ((tid & 31) == 0) { rL[wave] = accL; rP[wave] = accP; }
  __syncthreads();
  if (tid == 0) {
    float L = 0.0f, P = 0.0f;
#pragma unroll
    for (int wv = 0; wv < TPB / 32; ++wv) { L += rL[wv]; P += rP[wv]; }
    float invBF = 1.0f / ((float)rows * (float)FDIM);
    atomicAdd(out, L * invBF + P);   // global_atomic_add_f32
  }
}

extern "C" void kernel_launch(void* const* d_in, const int* in_sizes, int n_in,
                              void* d_out, int out_size, void* d_ws, size_t ws_size,
                              hipStream_t stream) {
  const float* in = (const float*)d_in[0];
  const int*   tg = (const int*)  d_in[1];
  const float* wt = (const float*)d_in[2];
  float*       out = (float*)d_out;

  int rows = in_sizes[0] / FDIM;          // B = 2,000,000
  (void)hipMemsetAsync(out, 0, sizeof(float), stream);   // graph-capturable memset node

  int grid = (rows + ROWS_PB - 1) / ROWS_PB;       // 1954 blocks of 8 wave32s
  WeightedMSELossWithPenalty_74028056314237_kernel<<<grid, TPB, 0, stream>>>(
      in, tg, wt, out, rows);
}